// NeuralSigKer_1400159338607
// MI455X (gfx1250) — compile-verified
//
#include <hip/hip_runtime.h>

// Neural signature kernel, fused per (x,y) pair:
//   phase 0: dX = diff(X[x]), dY = diff(Y[y]) into LDS (padded stride)
//   phase 1: 16-row panel of (<dx_s,dy_t> - 1) via v_wmma_f32_16x16x4_f32 (C init = -1)
//   phase 2: row recurrence new_row = 1 + cumsum(prev[t+1] + prev[t]*dm1) with
//            wave32 shuffle scan + cross-wave partials; coalesced row stores.

typedef float v2f __attribute__((ext_vector_type(2)));
typedef float v8f __attribute__((ext_vector_type(8)));

#define TT 256      // sequence length T
#define DD 16       // feature dim
#define DXS 20      // padded LDS row stride for increment panels (bank-conflict free)

__global__ __launch_bounds__(256) void sigker_wmma_kernel(
    const float* __restrict__ X, const float* __restrict__ Y,
    float* __restrict__ K)
{
    __shared__ float dxs[TT * DXS];   // dX rows (row 255 zero-padded)
    __shared__ float dys[TT * DXS];   // dY rows (row 255 zero-padded)
    __shared__ float dblk[16 * TT];   // (d - 1) for current 16-row s-panel
    __shared__ float prevrow[TT];     // K[s, :]
    __shared__ float partials[8];     // per-wave scan totals

    const int x    = blockIdx.x;
    const int y    = blockIdx.y;
    const int tid  = threadIdx.x;
    const int lane = tid & 31;
    const int wid  = tid >> 5;

    const float* Xb = X + (size_t)x * TT * DD;
    const float* Yb = Y + (size_t)y * TT * DD;
    float*       Kb = K + ((size_t)x * gridDim.y + y) * (size_t)TT * TT;

    // ---- Phase 0: increments into LDS ----
    for (int idx = tid; idx < (TT - 1) * DD; idx += 256) {
        int s = idx >> 4, k = idx & 15;
        dxs[s * DXS + k] = Xb[idx + DD] - Xb[idx];
        dys[s * DXS + k] = Yb[idx + DD] - Yb[idx];
    }
    if (tid < DD) {
        dxs[(TT - 1) * DXS + tid] = 0.0f;   // pad row s=255
        dys[(TT - 1) * DXS + tid] = 0.0f;
    }
    prevrow[tid] = 1.0f;                    // K[0, :] = var_0 = 1
    Kb[tid]      = 1.0f;                    // output row 0
    __syncthreads();

    const int row  = lane & 15;   // M (or N) index within tile
    const int half = lane >> 4;   // selects K-pair (fp32 A/B layout, ISA 7.12.2)

    for (int cblk = 0; cblk < 16; ++cblk) {
        // ---- Phase 1: dblk[sl][t] = <dx_{cblk*16+sl}, dy_t> - 1 via WMMA ----
        for (int ti = wid; ti < 16; ti += 8) {       // wave-uniform: EXEC all-1s
            v8f c;
            #pragma unroll
            for (int r = 0; r < 8; ++r) c[r] = -1.0f;   // fold the "-1" into C
            #pragma unroll
            for (int kc = 0; kc < 4; ++kc) {
                const int kbase = kc * 4 + half * 2;
                v2f a, b;
                a[0] = dxs[(cblk * 16 + row) * DXS + kbase];
                a[1] = dxs[(cblk * 16 + row) * DXS + kbase + 1];
                b[0] = dys[(ti   * 16 + row) * DXS + kbase];
                b[1] = dys[(ti   * 16 + row) * DXS + kbase + 1];
                // D = A(16x4) * B(4x16) + C, fp32 throughout
                c = __builtin_amdgcn_wmma_f32_16x16x4_f32(
                        false, a, false, b, (short)0, c, false, false);
            }
            #pragma unroll
            for (int r = 0; r < 8; ++r) {
                const int m = half * 8 + r;          // s_local of this element
                dblk[m * TT + ti * 16 + row] = c[r];
            }
        }
        __syncthreads();

        // ---- Phase 2: 16 recurrence rows (skip s=255 in last chunk) ----
        const int smax = (cblk == 15) ? 15 : 16;
        for (int sl = 0; sl < smax; ++sl) {
            const int s = cblk * 16 + sl;            // 0..254
            // c_t = prev[t+1] + prev[t]*(d-1), t<255; c_255 = 0
            float cv = 0.0f;
            if (tid < TT - 1)
                cv = prevrow[tid + 1] + prevrow[tid] * dblk[sl * TT + tid];

            // wave32 inclusive scan
            float v = cv;
            #pragma unroll
            for (int off = 1; off < 32; off <<= 1) {
                float n = __shfl_up(v, (unsigned)off, 32);
                if (lane >= off) v += n;
            }
            if (lane == 31) partials[wid] = v;
            __syncthreads();

            float offv = 0.0f;
            #pragma unroll
            for (int w = 0; w < 8; ++w)
                if (w < wid) offv += partials[w];
            const float newv = 1.0f + (v + offv);    // new_row[tid+1]

            float* Krow = Kb + (size_t)(s + 1) * TT;
            if (tid < TT - 1) {
                prevrow[tid + 1] = newv;             // prevrow[0] stays 1
                Krow[tid + 1]    = newv;
            } else {
                Krow[0] = 1.0f;                      // boundary column
            }
            __syncthreads();
        }
    }
}

extern "C" void kernel_launch(void* const* d_in, const int* in_sizes, int n_in,
                              void* d_out, int out_size, void* d_ws, size_t ws_size,
                              hipStream_t stream) {
    (void)n_in; (void)out_size; (void)d_ws; (void)ws_size;
    const float* X = (const float*)d_in[0];
    const float* Y = (const float*)d_in[1];
    float* K = (float*)d_out;

    const int Bx = in_sizes[0] / (TT * DD);   // 32
    const int By = in_sizes[1] / (TT * DD);   // 32

    dim3 grid(Bx, By);
    dim3 block(256);
    sigker_wmma_kernel<<<grid, block, 0, stream>>>(X, Y, K);
}